// RiRoIAlign_858993459360
// MI455X (gfx1250) — compile-verified
//
#include <hip/hip_runtime.h>
#include <math.h>

// ---------------------------------------------------------------------------
// RiRoIAlign for MI455X (gfx1250, wave32).
//
//   features: (B=2, Ctot=256, H=256, W=256) f32, addr = ((b*256+(c*8+q))*256+y)*256+x
//   rois:     (Rn=512, 6) = [batch, cx, cy, w, h, theta]
//   out:      (Rn, 32*8, 7, 7) f32
//
// Two kernels:
//   1) riroi_prep:  per-roi params (incl. hardware __sincosf) -> d_ws, once
//      per roi instead of 1568x per roi.
//   2) riroi_align_gather_kernel: one thread per (roi, channel, cell).
//      Sample coords depend only on (roi, cell), so the 4 bilinear corners
//      per sample are shared across all 8 orientation planes: accumulate
//      samp[q], q=0..7, then emit 8 orientation outputs as 2-FMA circular
//      blends -> 16 loads per output (half of naive). Gathers hit the 192 MB
//      L2 (feature tensor = 128 MB). No WMMA/TDM: no dense matmul exists,
//      and tile staging reads more than the samples it would serve.
// ---------------------------------------------------------------------------

#define RIROI_P       7
#define RIROI_CELLS   49            // 7*7
#define RIROI_O       8
#define RIROI_C       32
#define RIROI_W       256
#define RIROI_HW      65536         // 256*256
#define RIROI_BSTRIDE (256 * 65536) // elements per batch image
#define RIROI_INV2PI  0.15915494309189533577f
#define RIROI_NP      12            // floats of precomputed params per roi

// ---- per-ROI parameter precompute (512 threads total) ---------------------
__global__ __launch_bounds__(64) void riroi_prep(
    const float* __restrict__ rois,
    float* __restrict__ P,
    int Rn)
{
    const int r = blockIdx.x * 64 + threadIdx.x;
    if (r >= Rn) return;

    const float* roi = rois + r * 6;
    const float bf = roi[0];
    const float cw = roi[1] * 0.125f;
    const float ch = roi[2] * 0.125f;
    const float rw = fmaxf(roi[3] * 0.125f, 1.0f);
    const float rh = fmaxf(roi[4] * 0.125f, 1.0f);
    const float th = roi[5];

    const float ind_f     = (th * 8.0f) * RIROI_INV2PI;  // theta*O/(2*pi)
    const float ind_floor = floorf(ind_f);
    const float l_var     = ind_f - ind_floor;
    const float r_var     = 1.0f - l_var;
    const int   ind       = (((int)ind_floor % 8) + 8) % 8;

    // theta in [0, 2*pi): hardware v_sin_f32/v_cos_f32 are plenty accurate,
    // and avoid the branchy Payne-Hanek libm path in the hot kernel.
    float st, ct;
    __sincosf(th, &st, &ct);

    float* p = P + r * RIROI_NP;
    p[0]  = bf;
    p[1]  = cw;
    p[2]  = ch;
    p[3]  = rw;
    p[4]  = rh;
    p[5]  = ct;
    p[6]  = st;
    p[7]  = l_var;
    p[8]  = r_var;
    p[9]  = __int_as_float(ind);
    p[10] = rw / 7.0f;   // bin_w
    p[11] = rh / 7.0f;   // bin_h
}

// ---- main gather kernel ---------------------------------------------------
__global__ __launch_bounds__(256) void riroi_align_gather_kernel(
    const float* __restrict__ features,
    const float* __restrict__ P,
    float* __restrict__ out,
    int total_threads)  // Rn * 32 * 49
{
    const int tid = blockIdx.x * 256 + threadIdx.x;
    if (tid >= total_threads) return;

    const int cell = tid % RIROI_CELLS;          // lane-adjacent -> nearby pixels
    const int c    = (tid / RIROI_CELLS) % RIROI_C;
    const int r    = tid / (RIROI_CELLS * RIROI_C);
    const int px   = cell % RIROI_P;
    const int py   = cell / RIROI_P;

    // 3x global_load_b128, uniform per roi -> WGP$ hits.
    const float4* p4 = (const float4*)(P + r * RIROI_NP);
    const float4 pa = p4[0];   // bf, cw, ch, rw
    const float4 pb = p4[1];   // rh, ct, st, l_var
    const float4 pc = p4[2];   // r_var, ind(bits), bin_w, bin_h
    const int   b     = (int)pa.x;
    const float cw    = pa.y, ch = pa.z, rw = pa.w;
    const float rh    = pb.x, ct = pb.y, st = pb.z, l_var = pb.w;
    const float r_var = pc.x;
    const int   ind   = __float_as_int(pc.y);
    const float bin_w = pc.z, bin_h = pc.w;

    const float* fbase = features + (size_t)b * RIROI_BSTRIDE
                                  + (size_t)(c * RIROI_O) * RIROI_HW;

    float samp[RIROI_O];
    #pragma unroll
    for (int q = 0; q < RIROI_O; ++q) samp[q] = 0.0f;

    #pragma unroll
    for (int gy = 0; gy < 2; ++gy) {
        const float yy = -rh * 0.5f + ((float)py + (gy ? 0.75f : 0.25f)) * bin_h;
        #pragma unroll
        for (int gx = 0; gx < 2; ++gx) {
            const float xx = -rw * 0.5f + ((float)px + (gx ? 0.75f : 0.25f)) * bin_w;
            const float x = xx * ct - yy * st + cw;
            const float y = xx * st + yy * ct + ch;

            const bool valid = (y > -1.0f) && (y < 256.0f) &&
                               (x > -1.0f) && (x < 256.0f);
            const float yc = fmaxf(y, 0.0f);
            const float xc = fmaxf(x, 0.0f);
            int yl = (int)yc;
            int xl = (int)xc;
            const bool ye = yl >= 255;
            const bool xe = xl >= 255;
            const int yh = ye ? 255 : yl + 1;
            const int xh = xe ? 255 : xl + 1;
            yl = ye ? 255 : yl;
            xl = xe ? 255 : xl;
            const float yv = ye ? 255.0f : yc;
            const float xv = xe ? 255.0f : xc;
            const float ly = yv - (float)yl;
            const float lx = xv - (float)xl;
            const float hy = 1.0f - ly;
            const float hx = 1.0f - lx;
            const float vm = valid ? 1.0f : 0.0f;
            const float w00 = hy * hx * vm;
            const float w01 = hy * lx * vm;
            const float w10 = ly * hx * vm;
            const float w11 = ly * lx * vm;

            const int o00 = yl * RIROI_W + xl;
            const int o01 = yl * RIROI_W + xh;
            const int o10 = yh * RIROI_W + xl;
            const int o11 = yh * RIROI_W + xh;

            if (gy == 0 && gx == 0) {
                // gfx1250 global_prefetch_b8: warm the 8 plane rows of the
                // first sample while remaining weight math executes.
                #pragma unroll
                for (int q = 0; q < RIROI_O; ++q)
                    __builtin_prefetch(fbase + (size_t)q * RIROI_HW + o00, 0, 1);
            }

            // 32 independent global_load_b32 per sample point, unrolled so
            // ~32 loads stay in flight per loadcnt wait.
            #pragma unroll
            for (int q = 0; q < RIROI_O; ++q) {
                const float* fp = fbase + (size_t)q * RIROI_HW;
                samp[q] += fmaf(w00, fp[o00],
                           fmaf(w01, fp[o01],
                           fmaf(w10, fp[o10],
                                w11 * fp[o11])));
            }
        }
    }

    // out[o] = r_var*samp[(o-ind)%8] + l_var*samp[(o-ind+1)%8], mean /4.
    // Consecutive lanes (cells) -> consecutive store addresses per o.
    float* obase = out + ((size_t)(r * RIROI_C + c) * RIROI_O) * RIROI_CELLS + cell;
    #pragma unroll
    for (int o = 0; o < RIROI_O; ++o) {
        const int q0 = (o - ind + 8) & 7;
        const int q1 = (q0 + 1) & 7;
        obase[o * RIROI_CELLS] = fmaf(r_var, samp[q0], l_var * samp[q1]) * 0.25f;
    }
}

extern "C" void kernel_launch(void* const* d_in, const int* in_sizes, int n_in,
                              void* d_out, int out_size, void* d_ws, size_t ws_size,
                              hipStream_t stream) {
    const float* features = (const float*)d_in[0];
    const float* rois     = (const float*)d_in[1];
    float*       out      = (float*)d_out;
    float*       params   = (float*)d_ws;              // Rn*12 floats (24 KB)

    const int Rn    = in_sizes[1] / 6;                 // 512
    const int total = Rn * RIROI_C * RIROI_CELLS;      // each thread emits 8 outputs
    const int grid  = (total + 255) / 256;

    riroi_prep<<<(Rn + 63) / 64, 64, 0, stream>>>(rois, params, Rn);
    riroi_align_gather_kernel<<<grid, 256, 0, stream>>>(features, params, out, total);
}